// MultiEnhanceGroupFusionHead_65635690217909
// MI455X (gfx1250) — compile-verified
//
#include <hip/hip_runtime.h>

// ---- problem dims (fixed by the reference) ----
constexpr int ZN   = 64;     // batch
constexpr int ZC   = 256;    // template/search channels
constexpr int GCM  = 32;     // GC_MID
constexpr int H    = 64;     // search map H=W
constexpr int SPT  = 49;     // 7x7 taps
constexpr int NMAPS = ZN * ZC;           // 16384 maps
constexpr float OUT_SCALE  = 0.001f;
constexpr float POOL_SCALE = 1.0f / (float)(H * H);   // global avg pool

typedef __attribute__((ext_vector_type(2))) float v2f;
typedef __attribute__((ext_vector_type(8))) float v8f;

// ---------------------------------------------------------------------------
// K1: per (n,ci) 64x64 map -> 49 clipped rectangle sums S[i][j]
//   S(i,j) = sum of x over rows [max(0,i-3), min(63,i+60)] x cols likewise.
// b128 streaming: thread t owns float4 column slab 4*(t&15), row-group t>>4
// (rows == rg mod 4).  Iter 0 carries rows 0..3 (top edges), iter 15 rows
// 60..63 (bottom edges).  Partial column sums combine through LDS, then the
// column-clip corrections produce all 49 sums.  Reads x exactly once (268 MB
// total -> HBM roofline ~11.5us at 23.3 TB/s).
// ---------------------------------------------------------------------------
__global__ __launch_bounds__(64)
void k1_rect_sums(const float* __restrict__ x, float* __restrict__ S)
{
    __shared__ float lds_cs[4][64];    // per-row-group partial column sums
    __shared__ float lds_top[3][64];   // rows 0,1,2
    __shared__ float lds_bot[3][64];   // rows 61,62,63
    __shared__ float rs[7][64];        // row-clipped column sums
    __shared__ float tsum[7];

    const int m  = blockIdx.x;         // map = n*256 + ci
    const int t  = threadIdx.x;        // 0..63
    const int rg = t >> 4;             // row group: handles rows rg, rg+4, ...
    const int c4 = t & 15;             // float4 column index (cols 4*c4..+3)
    const float4* xp = (const float4*)(x + (size_t)m * (H * H));

    float4 acc   = {0.f, 0.f, 0.f, 0.f};
    float4 first = {0.f, 0.f, 0.f, 0.f};
    float4 last  = {0.f, 0.f, 0.f, 0.f};
#pragma unroll
    for (int it = 0; it < 16; ++it) {
        const float4 v = xp[(it * 4 + rg) * 16 + c4];   // global_load_b128
        if (it == 0)  first = v;                        // rows 0..3
        if (it == 15) last  = v;                        // rows 60..63
        acc.x += v.x; acc.y += v.y; acc.z += v.z; acc.w += v.w;
    }
    ((float4*)&lds_cs[rg][0])[c4] = acc;
    if (rg < 3)  ((float4*)&lds_top[rg][0])[c4]     = first;  // rows 0,1,2
    if (rg >= 1) ((float4*)&lds_bot[rg - 1][0])[c4] = last;   // rows 61,62,63
    __syncthreads();

    // thread t == column c from here on
    const float colsum = lds_cs[0][t] + lds_cs[1][t] + lds_cs[2][t] + lds_cs[3][t];
    const float t0 = lds_top[0][t], t1 = lds_top[1][t], t2 = lds_top[2][t];
    const float b0 = lds_bot[0][t], b1 = lds_bot[1][t], b2 = lds_bot[2][t];

    // RS[i][t]: column sum restricted to row-range of tap row i
    rs[0][t] = colsum - b0 - b1 - b2;   // rows 0..60
    rs[1][t] = colsum - b1 - b2;        // rows 0..61
    rs[2][t] = colsum - b2;             // rows 0..62
    rs[3][t] = colsum;                  // rows 0..63
    rs[4][t] = colsum - t0;             // rows 1..63
    rs[5][t] = colsum - t0 - t1;        // rows 2..63
    rs[6][t] = colsum - t0 - t1 - t2;   // rows 3..63
    __syncthreads();

    if (t < 7) {
        float a = 0.f;
#pragma unroll
        for (int c = 0; c < 64; ++c) a += rs[t][c];
        tsum[t] = a;
    }
    __syncthreads();

    if (t < SPT) {
        const int i = t / 7, j = t % 7;
        const float* r = &rs[i][0];
        float s = tsum[i];
        if      (j == 0) s -= r[61] + r[62] + r[63];  // cols 0..60
        else if (j == 1) s -= r[62] + r[63];          // cols 0..61
        else if (j == 2) s -= r[63];                  // cols 0..62
        else if (j == 4) s -= r[0];                   // cols 1..63
        else if (j == 5) s -= r[0] + r[1];            // cols 2..63
        else if (j == 6) s -= r[0] + r[1] + r[2];     // cols 3..63
        S[(size_t)m * SPT + t] = s;
    }
}

// ---------------------------------------------------------------------------
// K2: one wave per (n, c) with c = level*8 + g:
//   M32[n][c] = scale * sum_{k<32, ij<49} z_level[n, g*32+k, ij] * S[n, g*32+k, ij]
// Both operands are 1568 contiguous, 16B-aligned floats:
// 12 x b128 lane loads (512B per wave-load) + one b32 tail, then wave reduce.
// ---------------------------------------------------------------------------
__global__ __launch_bounds__(256)
void k2_template_dot(const float* __restrict__ z0, const float* __restrict__ z1,
                     const float* __restrict__ z2, const float* __restrict__ z3,
                     const float* __restrict__ S,  float* __restrict__ M32)
{
    const int wave = (blockIdx.x * blockDim.x + threadIdx.x) >> 5;  // 0..2047
    const int lane = threadIdx.x & 31;
    const int n = wave >> 5;           // 0..63
    const int c = wave & 31;           // 0..31
    const int level = c >> 3, g = c & 7;

    const float* z = (level == 0) ? z0 : (level == 1) ? z1 : (level == 2) ? z2 : z3;
    const size_t off = ((size_t)n * ZC + (size_t)g * GCM) * SPT;  // 16B-aligned slab
    const float* zp = z + off;
    const float* sp = S + off;

    float4 a4 = {0.f, 0.f, 0.f, 0.f};
#pragma unroll
    for (int it = 0; it < 12; ++it) {            // 12*128 = 1536 elements
        const int idx = it * 128 + lane * 4;
        const float4 zv = *(const float4*)(zp + idx);
        const float4 sv = *(const float4*)(sp + idx);
        a4.x = fmaf(zv.x, sv.x, a4.x);
        a4.y = fmaf(zv.y, sv.y, a4.y);
        a4.z = fmaf(zv.z, sv.z, a4.z);
        a4.w = fmaf(zv.w, sv.w, a4.w);
    }
    float acc = (a4.x + a4.y) + (a4.z + a4.w);
    {
        const int idx = 1536 + lane;             // tail: elements 1536..1567
        acc = fmaf(zp[idx], sp[idx], acc);
    }
#pragma unroll
    for (int msk = 16; msk >= 1; msk >>= 1) acc += __shfl_xor(acc, msk, 32);

    if (lane == 0)
        M32[n * 32 + c] = acc * (OUT_SCALE * POOL_SCALE);
}

// ---------------------------------------------------------------------------
// K3: out[n,o] = b[o] + sum_c M32[n,c] * W[o,c]
// Real GEMM (M=64, N=128, K=32) with V_WMMA_F32_16X16X4_F32 in fp32.
// One wave per 16x16 output tile; 8 K-steps of 4.  EXEC is all-ones.
// ---------------------------------------------------------------------------
__global__ __launch_bounds__(32)
void k3_mix_wmma(const float* __restrict__ M32, const float* __restrict__ W,
                 const float* __restrict__ bias, float* __restrict__ out)
{
    const int tile = blockIdx.x;          // 0..31
    const int n0 = (tile >> 3) * 16;      // 4 tiles along n
    const int o0 = (tile & 7) * 16;       // 8 tiles along o
    const int lane = threadIdx.x;
    const int half = lane >> 4;           // lanes 16-31 carry K+2/K+3
    const int l15  = lane & 15;

    v8f acc = {};
#pragma unroll
    for (int k = 0; k < 32; k += 4) {
        const int kk = k + 2 * half;
        const int arow = n0 + l15;
        v2f a; a.x = M32[arow * 32 + kk];     // A VGPR0
               a.y = M32[arow * 32 + kk + 1]; // A VGPR1
        const int bcol = o0 + l15;
        v2f b; b.x = W[bcol * 32 + kk];       // B row kk   across lanes
               b.y = W[bcol * 32 + kk + 1];   // B row kk+1 across lanes
        acc = __builtin_amdgcn_wmma_f32_16x16x4_f32(
            /*neg_a=*/false, a, /*neg_b=*/false, b,
            /*c_mod=*/(short)0, acc, /*reuse_a=*/false, /*reuse_b=*/false);
    }

    const int ocol = o0 + l15;
    const float bo = bias[ocol];
#pragma unroll
    for (int v = 0; v < 8; ++v) {
        const int nrow = n0 + v + 8 * half;   // C layout: VGPR v -> M = v (+8 hi half)
        out[nrow * 128 + ocol] = acc[v] + bo;
    }
}

// ---------------------------------------------------------------------------
extern "C" void kernel_launch(void* const* d_in, const int* in_sizes, int n_in,
                              void* d_out, int out_size, void* d_ws, size_t ws_size,
                              hipStream_t stream)
{
    (void)in_sizes; (void)n_in; (void)out_size; (void)ws_size;
    const float* z0 = (const float*)d_in[0];
    const float* z1 = (const float*)d_in[1];
    const float* z2 = (const float*)d_in[2];
    const float* z3 = (const float*)d_in[3];
    const float* x  = (const float*)d_in[4];
    const float* W  = (const float*)d_in[5];  // (128,32,1,1) -> [o*32+c]
    const float* b  = (const float*)d_in[6];  // (128,)
    float* out = (float*)d_out;               // (64,128) row-major

    float* S   = (float*)d_ws;                        // 16384*49 floats (~3.1 MB)
    float* M32 = S + (size_t)NMAPS * SPT;             // 64*32 floats

    k1_rect_sums   <<<NMAPS, 64,  0, stream>>>(x, S);
    k2_template_dot<<<256,   256, 0, stream>>>(z0, z1, z2, z3, S, M32);
    k3_mix_wmma    <<<32,    32,  0, stream>>>(M32, W, b, out);
}